// SCADecoder_68719476790
// MI455X (gfx1250) — compile-verified
//
#include <hip/hip_runtime.h>
#include <math.h>

typedef __attribute__((ext_vector_type(2))) float v2f;
typedef __attribute__((ext_vector_type(8))) float v8f;

namespace {
constexpr int Kc   = 512;
constexpr int Vc   = 50257;
constexpr int Dc   = 300;
constexpr int Rc   = 75;
constexpr int AUG  = 384;                      // 300 + 75 padded (zeros) to 384
constexpr int ITERS = 20;
constexpr int Vpad = ((Vc + 15) / 16) * 16;    // 50272
constexpr int NTILES = Vpad / 16;              // 3142 column tiles
constexpr int WPB  = 8;                        // waves per block in GEMM
constexpr int MT   = 4;                        // M-tiles (16 rows each) per wave
constexpr int NWPB = 4;                        // words per block in build_waug
}

// ---------------- block-wide sum reduction (blockDim.x == 256) --------------
__device__ __forceinline__ float block_sum_256(float v, float* red) {
  int t = threadIdx.x;
  __syncthreads();
  red[t] = v;
  __syncthreads();
  for (int off = 128; off > 0; off >>= 1) {
    if (t < off) red[t] += red[t + off];
    __syncthreads();
  }
  float r = red[0];
  __syncthreads();
  return r;
}

// ---------------- Kernel A: A' = [l2norm(a), l2norm(a)@U], na = ||A'||^2 ----
__global__ void build_aaug_kernel(const float* __restrict__ anchors,
                                  const float* __restrict__ MU,
                                  float* __restrict__ Aaug,
                                  float* __restrict__ na) {
  __shared__ float sA[Dc];
  __shared__ float red[256];
  int k = blockIdx.x;
  int t = threadIdx.x;
  float p = 0.f;
  for (int d = t; d < Dc; d += 256) {
    float x = anchors[(size_t)k * Dc + d];
    sA[d] = x;
    p += x * x;
  }
  float nn  = block_sum_256(p, red);
  float inv = 1.f / fmaxf(sqrtf(nn), 1e-12f);
  float p2 = 0.f;
  for (int d = t; d < Dc; d += 256) {
    float x = sA[d] * inv;
    sA[d] = x;
    Aaug[(size_t)k * AUG + d] = x;
    p2 += x * x;
  }
  float aa = block_sum_256(p2, red);   // barriers also make normalized sA visible
  float uu = 0.f;
  if (t < Rc) {
    float acc = 0.f;
    for (int d = 0; d < Dc; ++d) acc = fmaf(sA[d], MU[d * Rc + t], acc);
    Aaug[(size_t)k * AUG + Dc + t] = acc;
    uu = acc * acc;
  }
  if (t >= Rc && t < AUG - Dc) Aaug[(size_t)k * AUG + Dc + t] = 0.f;
  float uusum = block_sum_256(uu, red);
  if (t == 0) na[k] = aa + uusum;
}

// ------- Kernel B: w = l2norm(we @ Wp^T); W' = [w, w@U]; nw = ||W'||^2 ------
__global__ void build_waug_kernel(const float* __restrict__ we,
                                  const float* __restrict__ Wp,
                                  const float* __restrict__ MU,
                                  float* __restrict__ Waug,
                                  float* __restrict__ nw) {
  __shared__ float sIn[NWPB][Dc];
  __shared__ float sW[NWPB][Dc];
  __shared__ float red[256];
  int v0 = blockIdx.x * NWPB;
  int t  = threadIdx.x;

  for (int i = 0; i < NWPB; ++i) {
    int v = v0 + i;
    for (int d = t; d < Dc; d += 256)
      sIn[i][d] = (v < Vc) ? we[(size_t)v * Dc + d] : 0.f;
  }
  __syncthreads();

  // raw = we @ Wp^T : raw[d] = sum_e we[e] * Wp[d*Dc + e], 4 words at once
  float pp0 = 0.f, pp1 = 0.f, pp2 = 0.f, pp3 = 0.f;
  for (int d = t; d < Dc; d += 256) {
    const float* wr = Wp + (size_t)d * Dc;
    float a0 = 0.f, a1 = 0.f, a2 = 0.f, a3 = 0.f;
    for (int e = 0; e < Dc; ++e) {
      float w = wr[e];
      a0 = fmaf(sIn[0][e], w, a0);
      a1 = fmaf(sIn[1][e], w, a1);
      a2 = fmaf(sIn[2][e], w, a2);
      a3 = fmaf(sIn[3][e], w, a3);
    }
    sW[0][d] = a0; sW[1][d] = a1; sW[2][d] = a2; sW[3][d] = a3;
    pp0 += a0 * a0; pp1 += a1 * a1; pp2 += a2 * a2; pp3 += a3 * a3;
  }
  float pp[NWPB] = {pp0, pp1, pp2, pp3};

  for (int i = 0; i < NWPB; ++i) {
    int v = v0 + i;
    float nn  = block_sum_256(pp[i], red);
    float inv = 1.f / fmaxf(sqrtf(nn), 1e-12f);
    float p2 = 0.f;
    for (int d = t; d < Dc; d += 256) {
      float x = sW[i][d] * inv;
      sW[i][d] = x;
      if (v < Vc) Waug[(size_t)v * AUG + d] = x;
      p2 += x * x;
    }
    float ww = block_sum_256(p2, red);   // barriers make normalized sW visible
    float uu = 0.f;
    if (t < Rc) {
      float acc = 0.f;
      for (int d = 0; d < Dc; ++d) acc = fmaf(sW[i][d], MU[d * Rc + t], acc);
      if (v < Vc) Waug[(size_t)v * AUG + Dc + t] = acc;
      uu = acc * acc;
    }
    if (v < Vc && t >= Rc && t < AUG - Dc) Waug[(size_t)v * AUG + Dc + t] = 0.f;
    float uusum = block_sum_256(uu, red);
    if (v < Vc) {
      if (t == 0) nw[v] = ww + uusum;
    } else {
      for (int j = t; j < AUG; j += 256) Waug[(size_t)v * AUG + j] = 0.f;
      if (t == 0) nw[v] = 0.f;
    }
  }
}

// ------- Kernel C: log_kern[k,v] = -relu(na+nw-2*A'.W')/eps  (f32 WMMA) -----
// Each wave: one 16-col N tile x four 16-row M tiles (64 rows). B fragment is
// loaded once per k-step and reused across the 4 accumulators (4x less L2
// traffic on the 77MB W' matrix; A' is L2-resident).
__global__ void __launch_bounds__(256)
cost_wmma_kernel(const float* __restrict__ Aaug, const float* __restrict__ Waug,
                 const float* __restrict__ na, const float* __restrict__ nw,
                 const float* __restrict__ log_eps_p, float* __restrict__ out) {
  int wave  = threadIdx.x >> 5;
  int lane  = threadIdx.x & 31;
  int ntile = blockIdx.x * WPB + wave;
  if (ntile >= NTILES) return;               // wave-uniform exit
  int m0   = blockIdx.y * (16 * MT);         // 64-row band
  int n0   = ntile * 16;
  int half = lane >> 4;
  int l    = lane & 15;

  const float* aptr = Aaug + (size_t)(m0 + l) * AUG + 2 * half;
  const float* bptr = Waug + (size_t)(n0 + l) * AUG + 2 * half;

  v8f acc0 = {}, acc1 = {}, acc2 = {}, acc3 = {};
#pragma unroll 2
  for (int kk = 0; kk < AUG; kk += 4) {
    v2f bf = *(const v2f*)(bptr + kk);
    v2f a0 = *(const v2f*)(aptr + kk);
    v2f a1 = *(const v2f*)(aptr + 16 * AUG + kk);
    v2f a2 = *(const v2f*)(aptr + 32 * AUG + kk);
    v2f a3 = *(const v2f*)(aptr + 48 * AUG + kk);
    acc0 = __builtin_amdgcn_wmma_f32_16x16x4_f32(false, a0, false, bf,
                                                 (short)0, acc0, false, false);
    acc1 = __builtin_amdgcn_wmma_f32_16x16x4_f32(false, a1, false, bf,
                                                 (short)0, acc1, false, false);
    acc2 = __builtin_amdgcn_wmma_f32_16x16x4_f32(false, a2, false, bf,
                                                 (short)0, acc2, false, false);
    acc3 = __builtin_amdgcn_wmma_f32_16x16x4_f32(false, a3, false, bf,
                                                 (short)0, acc3, false, false);
  }

  float eps = log1pf(expf(*log_eps_p)) + 1e-4f;   // softplus(log_eps)+1e-4
  float neg_inv_eps = -1.f / eps;
  int n = n0 + l;
  if (n < Vc) {
    float nwv = nw[n];
    v8f accs[MT] = {acc0, acc1, acc2, acc3};
#pragma unroll
    for (int mt = 0; mt < MT; ++mt) {
#pragma unroll
      for (int j = 0; j < 8; ++j) {
        int m = m0 + mt * 16 + j + 8 * half;
        float cost = fmaxf(na[m] + nwv - 2.f * accs[mt][j], 0.f);
        out[(size_t)m * Vc + n] = cost * neg_inv_eps;
      }
    }
  }
}

// ---------------- Sinkhorn: init, row-LSE (log_u), col-LSE (log_v) ----------
__global__ void init_uv_kernel(float* __restrict__ log_u, float* __restrict__ log_v) {
  int i = blockIdx.x * 256 + threadIdx.x;
  if (i < Kc) log_u[i] = 0.f;
  if (i < Vc) log_v[i] = 0.f;
}

__global__ void row_update_kernel(const float* __restrict__ lk,
                                  const float* __restrict__ log_v,
                                  float* __restrict__ log_u) {
  __shared__ float redm[256];
  __shared__ float reds[256];
  int k = blockIdx.x, t = threadIdx.x;
  const float* row = lk + (size_t)k * Vc;
  float m = -INFINITY, s = 0.f;
  for (int v = t; v < Vc; v += 256) {
    float x = row[v] + log_v[v];
    if (x > m) { s *= expf(m - x); m = x; }
    s += expf(x - m);
  }
  redm[t] = m; reds[t] = s;
  __syncthreads();
  for (int off = 128; off > 0; off >>= 1) {
    if (t < off) {
      float m1 = redm[t], m2 = redm[t + off];
      float M = fmaxf(m1, m2);
      float sn = (m1 > -INFINITY ? reds[t]       * expf(m1 - M) : 0.f)
               + (m2 > -INFINITY ? reds[t + off] * expf(m2 - M) : 0.f);
      redm[t] = M; reds[t] = sn;
    }
    __syncthreads();
  }
  if (t == 0) log_u[k] = -(redm[0] + logf(reds[0]));
}

__global__ void col_update_kernel(const float* __restrict__ lk,
                                  const float* __restrict__ log_u,
                                  float* __restrict__ log_v) {
  __shared__ float su[Kc];
  int t = threadIdx.x;
  for (int k = t; k < Kc; k += 256) su[k] = log_u[k];
  __syncthreads();
  int v = blockIdx.x * 256 + t;
  if (v >= Vc) return;
  float m = -INFINITY, s = 0.f;
  for (int k = 0; k < Kc; ++k) {
    float x = lk[(size_t)k * Vc + v] + su[k];
    if (x > m) { s *= expf(m - x); m = x; }
    s += expf(x - m);
  }
  log_v[v] = -(m + logf(s));
}

// ---------------- Finalize: beta = softmax_row(log_P) (in place) ------------
__global__ void finalize_kernel(float* __restrict__ out,
                                const float* __restrict__ log_u,
                                const float* __restrict__ log_v,
                                const float* __restrict__ ts) {
  __shared__ float redm[256];
  __shared__ float reds[256];
  int k = blockIdx.x, t = threadIdx.x;
  float lu = log_u[k];
  float* row = out + (size_t)k * Vc;

  float m = -INFINITY, s = 0.f;
  for (int v = t; v < Vc; v += 256) {
    float x = row[v] + lu + log_v[v];
    if (x > m) { s *= expf(m - x); m = x; }
    s += expf(x - m);
  }
  redm[t] = m; reds[t] = s;
  __syncthreads();
  for (int off = 128; off > 0; off >>= 1) {
    if (t < off) {
      float m1 = redm[t], m2 = redm[t + off];
      float M = fmaxf(m1, m2);
      float sn = (m1 > -INFINITY ? reds[t]       * expf(m1 - M) : 0.f)
               + (m2 > -INFINITY ? reds[t + off] * expf(m2 - M) : 0.f);
      redm[t] = M; reds[t] = sn;
    }
    __syncthreads();
  }
  float L  = redm[0] + logf(reds[0]);
  float sp = log1pf(expf(ts[k]));            // softplus(topic_scale[k])

  float ssum = 0.f;
  for (int v = t; v < Vc; v += 256) {
    float x = row[v] + lu + log_v[v];
    float b = expf(x - L) * sp;
    row[v] = b;
    ssum += b;
  }
  float S2  = block_sum_256(ssum, reds);
  float inv = 1.f / S2;
  for (int v = t; v < Vc; v += 256) row[v] *= inv;
}

// ---------------------------------------------------------------------------
extern "C" void kernel_launch(void* const* d_in, const int* in_sizes, int n_in,
                              void* d_out, int out_size, void* d_ws, size_t ws_size,
                              hipStream_t stream) {
  const float* word_embeds = (const float*)d_in[0];   // [V, D]
  const float* anchors     = (const float*)d_in[1];   // [K, D]
  const float* MU          = (const float*)d_in[2];   // [D, R]
  const float* Wp          = (const float*)d_in[3];   // [D, D]
  const float* log_eps     = (const float*)d_in[4];   // scalar
  const float* topic_scale = (const float*)d_in[5];   // [K, 1]
  float* out = (float*)d_out;                         // [K, V] (log_kern, then beta)

  // workspace layout (floats): ~78 MB
  float* Waug  = (float*)d_ws;                        // [Vpad, AUG]
  float* Aaug  = Waug + (size_t)Vpad * AUG;           // [K, AUG]
  float* na    = Aaug + (size_t)Kc * AUG;             // [K]
  float* nw    = na + Kc;                             // [Vpad]
  float* log_u = nw + Vpad;                           // [K]
  float* log_v = log_u + Kc;                          // [Vpad]

  build_aaug_kernel<<<Kc, 256, 0, stream>>>(anchors, MU, Aaug, na);
  build_waug_kernel<<<Vpad / NWPB, 256, 0, stream>>>(word_embeds, Wp, MU, Waug, nw);

  dim3 g((NTILES + WPB - 1) / WPB, Kc / (16 * MT));
  cost_wmma_kernel<<<g, 256, 0, stream>>>(Aaug, Waug, na, nw, log_eps, out);

  init_uv_kernel<<<(Vc + 255) / 256, 256, 0, stream>>>(log_u, log_v);
  for (int it = 0; it < ITERS; ++it) {
    row_update_kernel<<<Kc, 256, 0, stream>>>(out, log_v, log_u);
    col_update_kernel<<<(Vc + 255) / 256, 256, 0, stream>>>(out, log_u, log_v);
  }
  finalize_kernel<<<Kc, 256, 0, stream>>>(out, log_u, log_v, topic_scale);
}